// SEModule_54924041781359
// MI455X (gfx1250) — compile-verified
//
#include <hip/hip_runtime.h>
#include <hip/hip_bf16.h>

// SE module: pool -> (WMMA f32 excite) -> gated scale.
// Shapes fixed by reference: N=32, C=256, Cm=64, H=W=56 (HW=3136).

typedef float v2f __attribute__((ext_vector_type(2)));
typedef float v4f __attribute__((ext_vector_type(4)));
typedef float v8f __attribute__((ext_vector_type(8)));

#define SE_N   32
#define SE_C   256
#define SE_CM  64
#define SE_HW  3136        // 56*56
#define SE_HW4 784         // HW / 4

// ---------------------------------------------------------------------------
// Kernel 1: global average pool. One 256-thread block per (n,c) plane.
// float4 loads keep us on the 23.3 TB/s HBM roofline; x lines land in L2
// (103 MB < 192 MB) so kernel 3's re-read hits cache.
// ---------------------------------------------------------------------------
__global__ void __launch_bounds__(256) se_pool(const float* __restrict__ x,
                                               float* __restrict__ s) {
    const int plane = blockIdx.x;                       // n*C + c
    const v4f* xp4 = (const v4f*)(x + (size_t)plane * SE_HW);

    float sum = 0.0f;
    for (int i = threadIdx.x; i < SE_HW4; i += 256) {
        v4f v = xp4[i];
        sum += v.x + v.y + v.z + v.w;
    }
    // wave32 reduction
    for (int off = 16; off > 0; off >>= 1)
        sum += __shfl_down(sum, off, 32);

    __shared__ float lds[8];
    if ((threadIdx.x & 31) == 0) lds[threadIdx.x >> 5] = sum;
    __syncthreads();
    if (threadIdx.x == 0) {
        float t = 0.0f;
#pragma unroll
        for (int w = 0; w < 8; ++w) t += lds[w];
        s[plane] = t * (1.0f / (float)SE_HW);
    }
}

// ---------------------------------------------------------------------------
// Kernel 2: excite. Single 256-thread block (8 waves), f32 WMMA 16x16x4.
//   GEMM1: h[32,64]  = relu(s[32,256] @ w1^T + b1)   -> 8 tiles, 1 per wave
//   GEMM2: g[32,256] = hsig(h[32,64]  @ w2^T + b2)   -> 32 tiles, 4 per wave
// A(16x4) per-lane: M = lane%16, elems K = {2*half, 2*half+1}, half = lane/16.
// B(4x16) symmetric: N = lane%16. C/D: row = r + 8*half, col = lane%16.
// EXEC is all ones everywhere around the WMMA ops (uniform control flow).
// ---------------------------------------------------------------------------
__global__ void __launch_bounds__(256) se_excite(const float* __restrict__ s,   // [32,256]
                                                 const float* __restrict__ w1,  // [64,256]
                                                 const float* __restrict__ b1,  // [64]
                                                 const float* __restrict__ w2,  // [256,64]
                                                 const float* __restrict__ b2,  // [256]
                                                 float* __restrict__ g) {       // [32,256]
    __shared__ float h[SE_N * SE_CM];                   // 8 KB

    const int lane = threadIdx.x & 31;
    const int wave = threadIdx.x >> 5;
    const int half = lane >> 4;
    const int lid  = lane & 15;

    // ---- GEMM1: one 16x16 tile per wave (2 M-tiles x 4 N-tiles) ----
    {
        const int m0 = (wave >> 2) * 16;                // batch rows
        const int n0 = (wave & 3) * 16;                 // Cm cols
        v8f acc = {};
        for (int k0 = 0; k0 < SE_C; k0 += 4) {
            const int ka = k0 + 2 * half;
            v2f a, b;
            a.x = s[(m0 + lid) * SE_C + ka];
            a.y = s[(m0 + lid) * SE_C + ka + 1];
            b.x = w1[(n0 + lid) * SE_C + ka];           // B[k][n] = w1[n][k]
            b.y = w1[(n0 + lid) * SE_C + ka + 1];
            acc = __builtin_amdgcn_wmma_f32_16x16x4_f32(
                false, a, false, b, (short)0, acc, false, false);
        }
        const float bias = b1[n0 + lid];                // per-column bias
#pragma unroll
        for (int r = 0; r < 8; ++r) {
            float v = acc[r] + bias;
            v = v > 0.0f ? v : 0.0f;                    // relu
            h[(m0 + r + 8 * half) * SE_CM + (n0 + lid)] = v;
        }
    }
    __syncthreads();

    // ---- GEMM2: 32 tiles (2 M-tiles x 16 N-tiles), 4 per wave ----
    for (int t = 0; t < 4; ++t) {
        const int idx = wave * 4 + t;
        const int m0  = (idx >> 4) * 16;
        const int n0  = (idx & 15) * 16;
        v8f acc = {};
        for (int k0 = 0; k0 < SE_CM; k0 += 4) {
            const int ka = k0 + 2 * half;
            v2f a, b;
            a.x = h[(m0 + lid) * SE_CM + ka];
            a.y = h[(m0 + lid) * SE_CM + ka + 1];
            b.x = w2[(n0 + lid) * SE_CM + ka];          // B[k][n] = w2[n][k]
            b.y = w2[(n0 + lid) * SE_CM + ka + 1];
            acc = __builtin_amdgcn_wmma_f32_16x16x4_f32(
                false, a, false, b, (short)0, acc, false, false);
        }
        const float bias = b2[n0 + lid];
#pragma unroll
        for (int r = 0; r < 8; ++r) {
            float v = acc[r] + bias;
            v = 1.2f * v + 3.0f;                        // hsigmoid
            v = v < 0.0f ? 0.0f : (v > 6.0f ? 6.0f : v);
            g[(m0 + r + 8 * half) * SE_C + (n0 + lid)] = v * (1.0f / 6.0f);
        }
    }
}

// ---------------------------------------------------------------------------
// Kernel 3: out = x * g (broadcast over HW). float4 per thread.
// Loads of x use default (RT) hint to hit the L2 lines left by se_pool;
// stores of out are nontemporal so 103 MB of output doesn't evict x mid-run.
// ---------------------------------------------------------------------------
__global__ void __launch_bounds__(256) se_scale(const float* __restrict__ x,
                                                const float* __restrict__ g,
                                                float* __restrict__ out) {
    const size_t i = (size_t)blockIdx.x * 256 + threadIdx.x;   // float4 index
    const int plane = (int)(i / SE_HW4);                       // n*C + c
    const float gv = g[plane];
    v4f v = ((const v4f*)x)[i];
    v4f o;
    o.x = v.x * gv; o.y = v.y * gv; o.z = v.z * gv; o.w = v.w * gv;
    __builtin_nontemporal_store(o, &((v4f*)out)[i]);
}

// ---------------------------------------------------------------------------
extern "C" void kernel_launch(void* const* d_in, const int* in_sizes, int n_in,
                              void* d_out, int out_size, void* d_ws, size_t ws_size,
                              hipStream_t stream) {
    const float* x  = (const float*)d_in[0];   // [32,256,56,56]
    const float* w1 = (const float*)d_in[1];   // [64,256]
    const float* b1 = (const float*)d_in[2];   // [64]
    const float* w2 = (const float*)d_in[3];   // [256,64]
    const float* b2 = (const float*)d_in[4];   // [256]
    float* out = (float*)d_out;

    float* s = (float*)d_ws;                   // [32,256]  = 32 KB
    float* g = s + SE_N * SE_C;                // [32,256]  = 32 KB

    se_pool<<<SE_N * SE_C, 256, 0, stream>>>(x, s);
    se_excite<<<1, 256, 0, stream>>>(s, w1, b1, w2, b2, g);

    const int n_vec4   = SE_N * SE_C * SE_HW4;          // 6,422,528
    const int n_blocks = n_vec4 / 256;                  // 25,088 (exact)
    se_scale<<<n_blocks, 256, 0, stream>>>(x, g, out);
}